// GeoGNNBlock_29068338659622
// MI455X (gfx1250) — compile-verified
//
#include <hip/hip_runtime.h>
#include <hip/hip_bf16.h>

#define N_NODES  100000
#define N_EDGES  800000
#define N_GRAPHS 4096
#define D   128
#define D2  256
#define LN_EPS 1e-5f

#define ROWS 64      // M rows per block in the fused MLP kernel
#define LDA  130     // LDS stride (floats) for A / H2 tile (pad avoids bank conflicts)
#define LDH  258     // LDS stride (floats) for H1 tile

typedef float v2f __attribute__((ext_vector_type(2)));
typedef float v8f __attribute__((ext_vector_type(8)));

// ---------------- workspace zeroing ----------------
__global__ void k_zero(float* __restrict__ p, int n) {
    int i = blockIdx.x * blockDim.x + threadIdx.x;
    int stride = gridDim.x * blockDim.x;
    for (; i < n; i += stride) p[i] = 0.0f;
}

// ---------------- graph node counts ----------------
__global__ void k_counts(const int* __restrict__ node2graph, float* __restrict__ counts) {
    int i = blockIdx.x * blockDim.x + threadIdx.x;
    if (i < N_NODES) atomicAdd(&counts[node2graph[i]], 1.0f);
}

// ---------------- edge message scatter-add ----------------
// One wave32 per edge; lane l handles floats [4l, 4l+4).
__global__ void k_scatter(const float4* __restrict__ nf, const float4* __restrict__ ef,
                          const int* __restrict__ src, const int* __restrict__ dst,
                          float* __restrict__ agg) {
    int e    = blockIdx.x * 8 + (threadIdx.x >> 5);
    int lane = threadIdx.x & 31;
    if (e >= N_EDGES) return;
    int s = src[e];
    int d = dst[e];
    float4 a = nf[(size_t)s * 32 + lane];
    float4 b = ef[(size_t)e * 32 + lane];
    float* out = agg + (size_t)d * D + lane * 4;
    atomicAdd(out + 0, a.x + b.x);
    atomicAdd(out + 1, a.y + b.y);
    atomicAdd(out + 2, a.z + b.z);
    atomicAdd(out + 3, a.w + b.w);
}

// ---------------- fused GIN MLP + LayerNorm + SqrtGraphNorm + ReLU + residual ----
// Block: 256 threads (8 waves). Block owns 64 rows, computes ALL 256 (H1) and
// 128 (H2) columns so LayerNorm can be done locally. H1 never touches HBM.
// Each wave owns an N-tile and sweeps all 4 M-tiles with 4 live accumulators:
// one B fragment feeds 4 WMMAs.
__launch_bounds__(256, 1)
__global__ void k_gin_mlp(const float* __restrict__ agg,
                          const float* __restrict__ W1, const float* __restrict__ b1,
                          const float* __restrict__ W2, const float* __restrict__ b2,
                          const float* __restrict__ gamma, const float* __restrict__ beta,
                          const float* __restrict__ counts, const int* __restrict__ n2g,
                          const float* __restrict__ node_feats,
                          float* __restrict__ out) {
    extern __shared__ float smem[];
    float* sA = smem;               // ROWS x LDA : A tile, later reused for H2
    float* sH = smem + ROWS * LDA;  // ROWS x LDH : H1 tile (ROWS x 256)

    const int tid  = threadIdx.x;
    const int wave = tid >> 5;
    const int lane = tid & 31;
    const int half = lane >> 4;   // 0: lanes 0-15, 1: lanes 16-31
    const int l16  = lane & 15;
    const int rowbase = blockIdx.x * ROWS;

    // ---- stage A tile (agg rows) into LDS, coalesced float4 loads ----
    for (int idx = tid; idx < ROWS * 32; idx += 256) {
        int r  = idx >> 5;
        int c4 = (idx & 31) * 4;
        int gr = rowbase + r;
        float4 v = make_float4(0.f, 0.f, 0.f, 0.f);
        if (gr < N_NODES) v = *(const float4*)(agg + (size_t)gr * D + c4);
        float* p = sA + r * LDA + c4;
        p[0] = v.x; p[1] = v.y; p[2] = v.z; p[3] = v.w;
    }
    __syncthreads();

    // ---- GEMM1: H1 = relu(A @ W1 + b1); A: 64x128, W1: 128x256 ----
    // 16 N-tiles / 8 waves = 2 per wave; each N-tile sweeps 4 M-tiles.
    for (int nt = wave; nt < 16; nt += 8) {
        const int nb = nt * 16;
        v8f acc0 = {}, acc1 = {}, acc2 = {}, acc3 = {};
        const float* ap = sA + l16 * LDA + 2 * half;          // + m*16*LDA + k0
        const float* bp = W1 + (size_t)(2 * half) * D2 + nb + l16;
        for (int k0 = 0; k0 < D; k0 += 4) {
            v2f b;
            b.x = bp[(size_t)k0 * D2];
            b.y = bp[(size_t)(k0 + 1) * D2];
            v2f a0; a0.x = ap[k0];               a0.y = ap[k0 + 1];
            v2f a1; a1.x = ap[16 * LDA + k0];    a1.y = ap[16 * LDA + k0 + 1];
            v2f a2; a2.x = ap[32 * LDA + k0];    a2.y = ap[32 * LDA + k0 + 1];
            v2f a3; a3.x = ap[48 * LDA + k0];    a3.y = ap[48 * LDA + k0 + 1];
            acc0 = __builtin_amdgcn_wmma_f32_16x16x4_f32(false, a0, false, b, (short)0, acc0, false, false);
            acc1 = __builtin_amdgcn_wmma_f32_16x16x4_f32(false, a1, false, b, (short)0, acc1, false, false);
            acc2 = __builtin_amdgcn_wmma_f32_16x16x4_f32(false, a2, false, b, (short)0, acc2, false, false);
            acc3 = __builtin_amdgcn_wmma_f32_16x16x4_f32(false, a3, false, b, (short)0, acc3, false, false);
        }
        float bias = b1[nb + l16];
        float* hp = sH + (8 * half) * LDH + nb + l16;
        #pragma unroll
        for (int i = 0; i < 8; ++i) {
            float h;
            h = acc0[i] + bias; hp[(i +  0) * LDH] = h > 0.f ? h : 0.f;
            h = acc1[i] + bias; hp[(i + 16) * LDH] = h > 0.f ? h : 0.f;
            h = acc2[i] + bias; hp[(i + 32) * LDH] = h > 0.f ? h : 0.f;
            h = acc3[i] + bias; hp[(i + 48) * LDH] = h > 0.f ? h : 0.f;
        }
    }
    __syncthreads();

    // ---- GEMM2: H2 = H1 @ W2 + b2; H1: 64x256, W2: 256x128 -> into sA ----
    // 8 N-tiles / 8 waves = 1 per wave; sweep 4 M-tiles.
    {
        const int nb = wave * 16;
        v8f acc0 = {}, acc1 = {}, acc2 = {}, acc3 = {};
        const float* ap = sH + l16 * LDH + 2 * half;
        const float* bp = W2 + (size_t)(2 * half) * D + nb + l16;
        for (int k0 = 0; k0 < D2; k0 += 4) {
            v2f b;
            b.x = bp[(size_t)k0 * D];
            b.y = bp[(size_t)(k0 + 1) * D];
            v2f a0; a0.x = ap[k0];               a0.y = ap[k0 + 1];
            v2f a1; a1.x = ap[16 * LDH + k0];    a1.y = ap[16 * LDH + k0 + 1];
            v2f a2; a2.x = ap[32 * LDH + k0];    a2.y = ap[32 * LDH + k0 + 1];
            v2f a3; a3.x = ap[48 * LDH + k0];    a3.y = ap[48 * LDH + k0 + 1];
            acc0 = __builtin_amdgcn_wmma_f32_16x16x4_f32(false, a0, false, b, (short)0, acc0, false, false);
            acc1 = __builtin_amdgcn_wmma_f32_16x16x4_f32(false, a1, false, b, (short)0, acc1, false, false);
            acc2 = __builtin_amdgcn_wmma_f32_16x16x4_f32(false, a2, false, b, (short)0, acc2, false, false);
            acc3 = __builtin_amdgcn_wmma_f32_16x16x4_f32(false, a3, false, b, (short)0, acc3, false, false);
        }
        float bias = b2[nb + l16];
        float* hp = sA + (8 * half) * LDA + nb + l16;
        #pragma unroll
        for (int i = 0; i < 8; ++i) {
            hp[(i +  0) * LDA] = acc0[i] + bias;
            hp[(i + 16) * LDA] = acc1[i] + bias;
            hp[(i + 32) * LDA] = acc2[i] + bias;
            hp[(i + 48) * LDA] = acc3[i] + bias;
        }
    }
    __syncthreads();

    // ---- epilogue: per-row LayerNorm + graphnorm + ReLU + residual ----
    // One wave per row; lane handles 4 floats; butterfly reduce for mean/var.
    for (int r = wave; r < ROWS; r += 8) {
        int gr = rowbase + r;
        const float* hrow = sA + r * LDA + lane * 4;
        float x0 = hrow[0], x1 = hrow[1], x2 = hrow[2], x3 = hrow[3];
        float s  = x0 + x1 + x2 + x3;
        float ss = x0 * x0 + x1 * x1 + x2 * x2 + x3 * x3;
        for (int off = 16; off >= 1; off >>= 1) {
            s  += __shfl_xor(s,  off, 32);
            ss += __shfl_xor(ss, off, 32);
        }
        float mu   = s * (1.0f / D);
        float var  = ss * (1.0f / D) - mu * mu;
        float rstd = rsqrtf(var + LN_EPS);
        if (gr < N_NODES) {
            float c    = counts[n2g[gr]];
            float invs = rsqrtf(c > 1.f ? c : 1.f);
            int c0 = lane * 4;
            const float* nfr = node_feats + (size_t)gr * D + c0;
            float* outr      = out        + (size_t)gr * D + c0;
            float4 res;
            float v;
            v = ((x0 - mu) * rstd * gamma[c0 + 0] + beta[c0 + 0]) * invs; v = v > 0.f ? v : 0.f; res.x = v + nfr[0];
            v = ((x1 - mu) * rstd * gamma[c0 + 1] + beta[c0 + 1]) * invs; v = v > 0.f ? v : 0.f; res.y = v + nfr[1];
            v = ((x2 - mu) * rstd * gamma[c0 + 2] + beta[c0 + 2]) * invs; v = v > 0.f ? v : 0.f; res.z = v + nfr[2];
            v = ((x3 - mu) * rstd * gamma[c0 + 3] + beta[c0 + 3]) * invs; v = v > 0.f ? v : 0.f; res.w = v + nfr[3];
            *(float4*)outr = res;
        }
    }
}

extern "C" void kernel_launch(void* const* d_in, const int* in_sizes, int n_in,
                              void* d_out, int out_size, void* d_ws, size_t ws_size,
                              hipStream_t stream) {
    const float* node_feats = (const float*)d_in[0];
    const float* edge_feats = (const float*)d_in[1];
    const int*   src        = (const int*)d_in[2];
    const int*   dst        = (const int*)d_in[3];
    const int*   n2g        = (const int*)d_in[4];
    const float* W1         = (const float*)d_in[5];
    const float* b1         = (const float*)d_in[6];
    const float* W2         = (const float*)d_in[7];
    const float* b2         = (const float*)d_in[8];
    const float* gamma      = (const float*)d_in[9];
    const float* beta       = (const float*)d_in[10];
    float* out = (float*)d_out;

    // workspace: agg [N_NODES*D] f32, then counts [N_GRAPHS] f32
    float* agg    = (float*)d_ws;
    float* counts = agg + (size_t)N_NODES * D;

    const int zn = N_NODES * D + N_GRAPHS;
    k_zero<<<2048, 256, 0, stream>>>(agg, zn);
    k_counts<<<(N_NODES + 255) / 256, 256, 0, stream>>>(n2g, counts);
    k_scatter<<<(N_EDGES + 7) / 8, 256, 0, stream>>>(
        (const float4*)node_feats, (const float4*)edge_feats, src, dst, agg);

    size_t shmem = (size_t)(ROWS * LDA + ROWS * LDH) * sizeof(float); // ~97 KB < 320 KB/WGP
    k_gin_mlp<<<(N_NODES + ROWS - 1) / ROWS, 256, shmem, stream>>>(
        agg, W1, b1, W2, b2, gamma, beta, counts, n2g, node_feats, out);
}